// BinaryTreeCRF_20409684591088
// MI455X (gfx1250) — compile-verified
//
#include <hip/hip_runtime.h>

typedef __attribute__((ext_vector_type(2))) float v2f;
typedef __attribute__((ext_vector_type(8))) float v8f;

#define LSTRIDE 34   // 32 + 2 pad: spreads odd/even rows across LDS banks

// ---------------------------------------------------------------------------
// Prep 1: per-label max of trans[k, :, :]  (k = 0..31, 1024 values each)
// ---------------------------------------------------------------------------
__global__ void bt_tmax_kernel(const float* __restrict__ trans,
                               float* __restrict__ tmax) {
    int k = threadIdx.x;
    if (k < 32) {
        const float* row = trans + k * 1024;
        float m = -__builtin_inff();
        for (int j = 0; j < 1024; ++j) m = fmaxf(m, row[j]);
        tmax[k] = m;
    }
}

// ---------------------------------------------------------------------------
// Prep 2: build the B-fragment stream for V_WMMA_F32_16X16X4_F32.
// B is [K=1024 x N=16] per k-tile t (t=0: k 0..15, t=1: k 16..31).
// Fragment layout per K-chunk c (K rows 4c..4c+3):
//   lanes 0-15 : VGPR0 = row 4c+0, VGPR1 = row 4c+1   (N = lane)
//   lanes 16-31: VGPR0 = row 4c+2, VGPR1 = row 4c+3   (N = lane-16)
// Stored so each lane does one global_load_b64 per (t, c):
//   frag[ ((t*256 + c)*32 + lane)*2 + v ]
// Value = exp(trans[k][K] - tmax[k]),  K = (l,r) flattened = l*32 + r.
// ---------------------------------------------------------------------------
__global__ void bt_frag_kernel(const float* __restrict__ trans,
                               const float* __restrict__ tmax,
                               float* __restrict__ frag) {
    int idx = blockIdx.x * blockDim.x + threadIdx.x;   // 0..32767
    int v     = idx & 1;
    int pair  = idx >> 1;
    int laneF = pair & 31;
    int c     = (pair >> 5) & 255;
    int t     = pair >> 13;
    int K = 4 * c + ((laneF >> 4) << 1) + v;
    int k = t * 16 + (laneF & 15);
    frag[idx] = __expf(trans[k * 1024 + K] - tmax[k]);
}

// ---------------------------------------------------------------------------
// One tree level: n internal nodes from 2n child score rows.
// Per wave: 16-node tile.
//   D[node][k] = O @ ExpT^T via 256 K-chunks of V_WMMA_F32_16X16X4_F32 (x2 k-tiles)
//   A fragment built on the fly: A[node][l*32+r] = a_node[l] * b_node[r]
// ---------------------------------------------------------------------------
__global__ void bt_level_kernel(const float* __restrict__ src,   // [2n x 32]
                                float* __restrict__ dst,         // [n x 32]
                                const float* __restrict__ em,    // [n x 32]
                                const float* __restrict__ frag,  // [2*256*32*2]
                                const float* __restrict__ tmax,  // [32]
                                int n) {
    __shared__ float ab[4][32 * LSTRIDE];  // exp'd child rows, per wave
    __shared__ float mx[4][32];            // per-child-row max, per wave

    const int lane = threadIdx.x & 31;
    const int wave = threadIdx.x >> 5;
    const int numTiles = (n + 15) >> 4;
    int tile = blockIdx.x * 4 + wave;
    const bool active = tile < numTiles;
    if (!active) tile = numTiles - 1;      // clamp: all waves do identical work

    // ---- Phase 1: each lane owns one child row (32 floats): max + exp -> LDS
    {
        int row = tile * 32 + lane;                    // global child row
        int rowc = row < 2 * n ? row : 2 * n - 1;      // clamp for tail tiles
        const float* p = src + rowc * 32;
        float vals[32];
        float m = -__builtin_inff();
        #pragma unroll
        for (int j = 0; j < 32; ++j) { vals[j] = p[j]; m = fmaxf(m, vals[j]); }
        mx[wave][lane] = m;
        #pragma unroll
        for (int j = 0; j < 32; ++j)
            ab[wave][lane * LSTRIDE + j] = __expf(vals[j] - m);
    }
    __syncthreads();

    // ---- Phase 2: WMMA accumulation over K = 1024 in chunks of 4
    const int i16 = lane >> 4;     // which half of the wave
    const int il  = lane & 15;     // node-in-tile (A rows) / k-in-tile (B cols)
    const float* aRow = &ab[wave][(2 * il) * LSTRIDE];       // left  = even row
    const float* bRow = &ab[wave][(2 * il + 1) * LSTRIDE];   // right = odd row
    const float* fragBase = frag + lane * 2;

    v8f acc0 = {0.f, 0.f, 0.f, 0.f, 0.f, 0.f, 0.f, 0.f};
    v8f acc1 = acc0;

    for (int l = 0; l < 32; ++l) {          // a[l] constant over 8 chunks
        float av = aRow[l];
        #pragma unroll
        for (int cc = 0; cc < 8; ++cc) {
            int c  = l * 8 + cc;            // K-chunk index, K rows 4c..4c+3
            int r0 = 4 * cc + 2 * i16;      // r for this lane's VGPR0
            v2f bb = *reinterpret_cast<const v2f*>(&bRow[r0]);   // ds_load_b64
            v2f afrag;
            afrag.x = av * bb.x;            // A[K=4c+2*i16  ] for M=il
            afrag.y = av * bb.y;            // A[K=4c+2*i16+1]
            v2f b0 = *reinterpret_cast<const v2f*>(fragBase + c * 64);
            v2f b1 = *reinterpret_cast<const v2f*>(fragBase + (256 + c) * 64);
            acc0 = __builtin_amdgcn_wmma_f32_16x16x4_f32(
                false, afrag, false, b0, (short)0, acc0, false, false);
            acc1 = __builtin_amdgcn_wmma_f32_16x16x4_f32(
                false, afrag, false, b1, (short)0, acc1, false, false);
        }
    }

    // ---- Phase 3: epilogue. C/D layout: VGPR i -> (M = i + 8*i16, N = il)
    float tmk0 = tmax[il];
    float tmk1 = tmax[16 + il];
    #pragma unroll
    for (int i = 0; i < 8; ++i) {
        int localNode = i + 8 * i16;
        int node = tile * 16 + localNode;
        if (active && node < n) {
            float mL = mx[wave][2 * localNode];
            float mR = mx[wave][2 * localNode + 1];
            float base = mL + mR;
            dst[node * 32 + il]      = em[node * 32 + il]      + base + tmk0 + __logf(acc0[i]);
            dst[node * 32 + 16 + il] = em[node * 32 + 16 + il] + base + tmk1 + __logf(acc1[i]);
        }
    }
}

// ---------------------------------------------------------------------------
extern "C" void kernel_launch(void* const* d_in, const int* in_sizes, int n_in,
                              void* d_out, int out_size, void* d_ws, size_t ws_size,
                              hipStream_t stream) {
    const float* leaf     = (const float*)d_in[0];   // [8192 x 32]
    const float* internal = (const float*)d_in[1];   // [8191 x 32]
    const float* trans    = (const float*)d_in[2];   // [32 x 32 x 32]

    float* frag = (float*)d_ws;            // 32768 floats (128 KB)
    float* tmax = frag + 32768;            // 32 floats
    float* buf0 = tmax + 32;               // 4096*32 floats
    float* buf1 = buf0 + 4096 * 32;        // 4096*32 floats

    bt_tmax_kernel<<<1, 32, 0, stream>>>(trans, tmax);
    bt_frag_kernel<<<128, 256, 0, stream>>>(trans, tmax, frag);

    const float* src = leaf;
    int off = 0;
    int flip = 0;
    for (int n = 4096; n >= 1; n >>= 1) {
        float* dst = (n == 1) ? (float*)d_out : (flip ? buf1 : buf0);
        int numTiles = (n + 15) >> 4;
        int blocks = (numTiles + 3) >> 2;          // 4 waves (tiles) per block
        bt_level_kernel<<<blocks, 128, 0, stream>>>(src, dst,
                                                    internal + (size_t)off * 32,
                                                    frag, tmax, n);
        src = dst;
        off += n;
        flip ^= 1;
    }
}